// SegmentAggregation_1219770712104
// MI455X (gfx1250) — compile-verified
//
#include <hip/hip_runtime.h>

// Segment-sum (sorted ids), two-path CDNA5 implementation:
//  - Bulk path: uniform 16-row windows (segid[first]==segid[last] => whole
//    window is one segment) streamed via global_load_async_to_lds_b128 with
//    double-buffered LDS staging (ASYNCcnt pipelining), run-length float4
//    accumulation, rare f32-atomic flushes.
//  - Boundary path: windows containing a segment boundary go through the
//    one-hot GEMM formulation on the matrix pipe (V_WMMA_F32_16X16X4_F32).
// Roofline: ~1.03 GB read + ~12 MB write/atomics @ 23.3 TB/s => ~45 us floor.

#define DIM             128
#define WIN             16            // rows per window
#define CHUNK_WINS      32            // windows per stream-wave (512 rows)
#define WAVES_PER_BLOCK 4

typedef float v2f __attribute__((ext_vector_type(2)));
typedef float v8f __attribute__((ext_vector_type(8)));

__global__ void seg_zero_kernel(float4* out4, int n4) {
    int i = blockIdx.x * blockDim.x + threadIdx.x;
    if (i < n4) out4[i] = make_float4(0.f, 0.f, 0.f, 0.f);
}

// ---------------------------------------------------------------------------
// Bulk streaming kernel: async-DMA global->LDS, float4 run-length accumulate.
// Wave covers all 128 columns (lane -> 4 consecutive cols).
// ---------------------------------------------------------------------------
__global__ __launch_bounds__(32 * WAVES_PER_BLOCK)
void seg_stream_kernel(const float* __restrict__ data,
                       const int* __restrict__ segid,
                       float* __restrict__ out,
                       int nrows) {
    // 2 buffers x 16 rows x 128 floats per wave = 16KB/wave, 64KB/block
    __shared__ __align__(16) float stage[WAVES_PER_BLOCK * 2 * WIN * DIM];

    const int lane  = threadIdx.x & 31;
    const int wid   = threadIdx.x >> 5;
    const int chunk = blockIdx.x * WAVES_PER_BLOCK + wid;
    const int row0  = chunk * (WIN * CHUNK_WINS);
    if (row0 >= nrows) return;                      // wave-uniform exit
    const int rend  = min(row0 + WIN * CHUNK_WINS, nrows);
    const int nfull = (rend - row0) >> 4;           // full windows only
    if (nfull == 0) return;

    const int stbase = wid * (2 * WIN * DIM);       // float index of wave's slab
    // LDS byte address for async DMA: low 32 bits of the flat shared address
    // are the wave-relative LDS offset (flat LDS aperture truncation).
    const unsigned lds0 = (unsigned)(uintptr_t)&stage[stbase] + (unsigned)(lane * 16);
    const unsigned lds1 = lds0 + WIN * DIM * 4;     // second buffer, 8KB apart

    // Prefetch window 0 into buffer 0. One instruction per row: the shared
    // 24-bit offset advances both the global and LDS addresses by 512B.
    {
        const float* g = data + (size_t)row0 * DIM + lane * 4;
        #pragma unroll
        for (int k = 0; k < WIN; ++k)
            asm volatile("global_load_async_to_lds_b128 %0, %1, off offset:%2"
                         :: "v"(lds0), "v"(g), "i"(k * DIM * 4) : "memory");
    }

    float4 acc = make_float4(0.f, 0.f, 0.f, 0.f);
    int cur = -1;                                   // current run's segment id

    for (int w = 0; w < nfull; ++w) {
        const int rw = row0 + w * WIN;
        const int iF = segid[rw];                   // wave-uniform broadcasts
        const int iL = segid[rw + WIN - 1];

        // Prior window's LDS reads must retire before overwriting that buffer.
        asm volatile("s_wait_dscnt 0x0" ::: "memory");
        if (w + 1 < nfull) {
            const unsigned dst = ((w + 1) & 1) ? lds1 : lds0;
            const float* g = data + (size_t)(rw + WIN) * DIM + lane * 4;
            #pragma unroll
            for (int k = 0; k < WIN; ++k)
                asm volatile("global_load_async_to_lds_b128 %0, %1, off offset:%2"
                             :: "v"(dst), "v"(g), "i"(k * DIM * 4) : "memory");
            // 16 newer loads outstanding; async loads complete in order, so
            // cnt<=16 means window w has fully landed in LDS.
            asm volatile("s_wait_asynccnt 0x10" ::: "memory");
        } else {
            asm volatile("s_wait_asynccnt 0x0" ::: "memory");
        }

        const int bufoff = stbase + ((w & 1) ? WIN * DIM : 0) + lane * 4;
        if (iF == iL) {
            // Uniform window: pure run-length accumulate at full bandwidth.
            if (iF != cur) {
                if (cur >= 0) {
                    float* o = &out[(size_t)cur * DIM + lane * 4];
                    atomicAdd(o + 0, acc.x); atomicAdd(o + 1, acc.y);
                    atomicAdd(o + 2, acc.z); atomicAdd(o + 3, acc.w);
                }
                acc = make_float4(0.f, 0.f, 0.f, 0.f);
                cur = iF;
            }
            #pragma unroll
            for (int k = 0; k < WIN; ++k) {
                const float4 v =
                    *reinterpret_cast<const float4*>(&stage[bufoff + k * DIM]);
                acc.x += v.x; acc.y += v.y; acc.z += v.z; acc.w += v.w;
            }
        } else {
            // Boundary window: owned by the WMMA kernel; run continuity broken.
            if (cur >= 0) {
                float* o = &out[(size_t)cur * DIM + lane * 4];
                atomicAdd(o + 0, acc.x); atomicAdd(o + 1, acc.y);
                atomicAdd(o + 2, acc.z); atomicAdd(o + 3, acc.w);
            }
            acc = make_float4(0.f, 0.f, 0.f, 0.f);
            cur = -1;
        }
    }
    if (cur >= 0) {
        float* o = &out[(size_t)cur * DIM + lane * 4];
        atomicAdd(o + 0, acc.x); atomicAdd(o + 1, acc.y);
        atomicAdd(o + 2, acc.z); atomicAdd(o + 3, acc.w);
    }
}

// ---------------------------------------------------------------------------
// Boundary kernel: one wave per 16-row window; exits unless the window spans a
// segment boundary. One-hot A (16x4) x data B (4x16) on the matrix pipe,
// 8 column slabs => 32 v_wmma_f32_16x16x4_f32 per boundary window.
// ---------------------------------------------------------------------------
__global__ __launch_bounds__(32 * WAVES_PER_BLOCK)
void seg_boundary_wmma_kernel(const float* __restrict__ data,
                              const int* __restrict__ segid,
                              float* __restrict__ out,
                              int nrows) {
    __shared__ int map_lds[WAVES_PER_BLOCK * 16];

    const int lane = threadIdx.x & 31;
    const int wid  = threadIdx.x >> 5;
    const int win  = blockIdx.x * WAVES_PER_BLOCK + wid;
    const int nwin_full = nrows >> 4;
    const int total = nwin_full + ((nrows & 15) ? 1 : 0);
    if (win >= total) return;
    const int r0 = win << 4;

    if (win < nwin_full) {
        if (segid[r0] == segid[r0 + WIN - 1]) return;   // uniform: bulk kernel's

        const int  m  = lane & 15;
        const bool hi = lane >= 16;
        int* map = &map_lds[wid * 16];

        v8f acc[8];
        #pragma unroll
        for (int s = 0; s < 8; ++s)
            #pragma unroll
            for (int v = 0; v < 8; ++v) acc[s][v] = 0.0f;

        int count = 0, last = -1;       // <=16 distinct ids in 16 sorted rows
        #pragma unroll
        for (int g = 0; g < 4; ++g) {
            const int r  = r0 + g * 4;
            const int i0 = segid[r + 0], i1 = segid[r + 1];
            const int i2 = segid[r + 2], i3 = segid[r + 3];
            const int c0 = count + (i0 != last);
            const int c1 = c0 + (i1 != i0);
            const int c2 = c1 + (i2 != i1);
            const int c3 = c2 + (i3 != i2);
            const int r0k = c0 - 1, r1k = c1 - 1, r2k = c2 - 1, r3k = c3 - 1;
            count = c3; last = i3;
            if (lane == 0) { map[r0k] = i0; map[r1k] = i1; map[r2k] = i2; map[r3k] = i3; }

            // A 16x4 f32 layout: VGPR0 = {K0 | K2}, VGPR1 = {K1 | K3}
            v2f a;
            a.x = (m == (hi ? r2k : r0k)) ? 1.0f : 0.0f;
            a.y = (m == (hi ? r3k : r1k)) ? 1.0f : 0.0f;

            const float* rp = data + (size_t)r * DIM + m;
            #pragma unroll
            for (int s = 0; s < 8; ++s) {
                v2f b;      // B 4x16: low lanes K rows 0..1, high lanes 2..3
                b.x = rp[s * 16 + (hi ? 2 : 0) * DIM];
                b.y = rp[s * 16 + (hi ? 3 : 1) * DIM];
                acc[s] = __builtin_amdgcn_wmma_f32_16x16x4_f32(
                    false, a, false, b, (short)0, acc[s], false, false);
            }
        }

        asm volatile("s_wait_dscnt 0x0" ::: "memory");
        #pragma unroll
        for (int v = 0; v < 8; ++v) {
            const int M = v + (hi ? 8 : 0);     // C/D: lanes0-15 M=v, 16-31 M=v+8
            if (M < count) {
                const size_t so = (size_t)map[M] * DIM + m;
                #pragma unroll
                for (int s = 0; s < 8; ++s)
                    atomicAdd(&out[so + s * 16], acc[s][v]);
            }
        }
    } else {
        // Scalar tail rows (N % 16 != 0); lane covers 4 consecutive columns.
        for (int r = r0; r < nrows; ++r) {
            const int s = segid[r];
            const float* rp = data + (size_t)r * DIM + lane * 4;
            float* o = &out[(size_t)s * DIM + lane * 4];
            atomicAdd(o + 0, rp[0]); atomicAdd(o + 1, rp[1]);
            atomicAdd(o + 2, rp[2]); atomicAdd(o + 3, rp[3]);
        }
    }
}

extern "C" void kernel_launch(void* const* d_in, const int* in_sizes, int n_in,
                              void* d_out, int out_size, void* d_ws, size_t ws_size,
                              hipStream_t stream) {
    const float* data = (const float*)d_in[0];
    const int*   seg  = (const int*)d_in[1];
    float*       out  = (float*)d_out;

    const int nrows = in_sizes[0] / DIM;

    // 1) Zero output (harness poisons d_out).
    const int n4 = out_size / 4;
    seg_zero_kernel<<<(n4 + 255) / 256, 256, 0, stream>>>((float4*)out, n4);

    // 2) Bulk streaming path (uniform windows).
    const int rows_per_chunk = WIN * CHUNK_WINS;
    const int chunks  = (nrows + rows_per_chunk - 1) / rows_per_chunk;
    const int blocksA = (chunks + WAVES_PER_BLOCK - 1) / WAVES_PER_BLOCK;
    seg_stream_kernel<<<blocksA, 32 * WAVES_PER_BLOCK, 0, stream>>>(data, seg, out, nrows);

    // 3) Boundary windows on the matrix pipe (+ scalar tail).
    const int nwin_full = nrows >> 4;
    const int total     = nwin_full + ((nrows & 15) ? 1 : 0);
    const int blocksB   = (total + WAVES_PER_BLOCK - 1) / WAVES_PER_BLOCK;
    seg_boundary_wmma_kernel<<<blocksB, 32 * WAVES_PER_BLOCK, 0, stream>>>(data, seg, out, nrows);
}